// Attention_46858093199829
// MI455X (gfx1250) — compile-verified
//
#include <hip/hip_runtime.h>
#include <hip/hip_bf16.h>

typedef float v2f __attribute__((ext_vector_type(2)));
typedef float v8f __attribute__((ext_vector_type(8)));

// Phase 1: single workgroup (8 wave32) computes
//   sk = key @ wk  via v_wmma_f32_16x16x4_f32 (16 rows of key per WMMA),
//   p  = softmax(sk),  o[d] = sum_j p[j]*value[j][d]   -> 3 floats in ws.
__global__ __launch_bounds__(256) void attn_reduce_kernel(
    const float* __restrict__ key, const float* __restrict__ value,
    const float* __restrict__ W, float* __restrict__ o)
{
    __shared__ float sk[2048];
    __shared__ float red[256];
    __shared__ float red4[4][256];

    const int tid  = threadIdx.x;
    const int wave = tid >> 5;
    const int ln   = tid & 31;
    const int row  = ln & 15;
    const bool lo  = (ln < 16);

    // wk = W[0, 3:6]
    const float wk0 = W[3], wk1 = W[4], wk2 = W[5];

    // B matrix (4x16 f32): every column n holds wk (K=3 padded with 0).
    // Layout mirror of A: lanes 0-15 -> K=0,1 in v0,v1; lanes 16-31 -> K=2,3.
    v2f bmat;
    bmat.x = lo ? wk0 : wk2;
    bmat.y = lo ? wk1 : 0.0f;

    // Each wave produces sk for 256 consecutive j's, 16 per WMMA.
    for (int c = 0; c < 16; ++c) {
        const int j0 = wave * 256 + c * 16;
        const int j  = j0 + row;
        // A matrix (16x4 f32): row m = key[j0+m][0..2], K=3 padded with 0.
        v2f amat;
        amat.x = lo ? key[j * 3 + 0] : key[j * 3 + 2];
        amat.y = lo ? key[j * 3 + 1] : 0.0f;
        v8f acc = {};
        acc = __builtin_amdgcn_wmma_f32_16x16x4_f32(
            /*neg_a=*/false, amat, /*neg_b=*/false, bmat,
            /*c_mod=*/(short)0, acc, /*reuse_a=*/false, /*reuse_b=*/false);
        // D layout: VGPR r, lanes 0-15 -> M=r, lanes 16-31 -> M=r+8 (all columns equal).
        if (row == 0) {
            const int mbase = lo ? 0 : 8;
            #pragma unroll
            for (int r = 0; r < 8; ++r) sk[j0 + mbase + r] = acc[r];
        }
    }
    __syncthreads();

    // Global max of sk (softmax shift is exact: the per-(b,i) terms cancel).
    float m = -3.402823466e38f;
    #pragma unroll
    for (int k = 0; k < 8; ++k) m = fmaxf(m, sk[tid + k * 256]);
    red[tid] = m;
    __syncthreads();
    for (int s = 128; s > 0; s >>= 1) {
        if (tid < s) red[tid] = fmaxf(red[tid], red[tid + s]);
        __syncthreads();
    }
    const float gmax = red[0];
    __syncthreads();

    // e = exp(sk - max); accumulate denominator and e-weighted value sums.
    float den = 0.0f, n0 = 0.0f, n1 = 0.0f, n2 = 0.0f;
    #pragma unroll
    for (int k = 0; k < 8; ++k) {
        const int j = tid + k * 256;
        const float e = expf(sk[j] - gmax);
        den += e;
        n0  += e * value[j * 3 + 0];
        n1  += e * value[j * 3 + 1];
        n2  += e * value[j * 3 + 2];
    }
    red4[0][tid] = den; red4[1][tid] = n0; red4[2][tid] = n1; red4[3][tid] = n2;
    __syncthreads();
    for (int s = 128; s > 0; s >>= 1) {
        if (tid < s) {
            red4[0][tid] += red4[0][tid + s];
            red4[1][tid] += red4[1][tid + s];
            red4[2][tid] += red4[2][tid + s];
            red4[3][tid] += red4[3][tid + s];
        }
        __syncthreads();
    }
    if (tid == 0) {
        const float inv = 1.0f / red4[0][0];
        o[0] = red4[1][0] * inv;
        o[1] = red4[2][0] * inv;
        o[2] = red4[3][0] * inv;
    }
}

// Phase 2: broadcast the 3-vector into out[32][512][3] with float4 stores.
// Group g covers elements 4g..4g+3; their d-indices are ((4g+t) % 3), and
// (4g) % 3 == g % 3, so only 3 distinct float4 patterns exist.
__global__ __launch_bounds__(256) void attn_bcast_kernel(
    const float* __restrict__ o, float4* __restrict__ out, int ngroups)
{
    const int g = blockIdx.x * blockDim.x + threadIdx.x;
    if (g >= ngroups) return;
    const float v0 = o[0], v1 = o[1], v2 = o[2];
    const int r = g % 3;
    const float a = (r == 0) ? v0 : (r == 1) ? v1 : v2;
    const float b = (r == 0) ? v1 : (r == 1) ? v2 : v0;
    const float c = (r == 0) ? v2 : (r == 1) ? v0 : v1;
    out[g] = make_float4(a, b, c, a);
}

extern "C" void kernel_launch(void* const* d_in, const int* in_sizes, int n_in,
                              void* d_out, int out_size, void* d_ws, size_t ws_size,
                              hipStream_t stream) {
    // inputs: 0=x (unused: cancels in softmax), 1=key, 2=value, 3=W, 4=b (unused)
    const float* key   = (const float*)d_in[1];
    const float* value = (const float*)d_in[2];
    const float* W     = (const float*)d_in[3];
    float* o = (float*)d_ws;                 // 3 floats of scratch

    attn_reduce_kernel<<<1, 256, 0, stream>>>(key, value, W, o);

    const int ngroups = out_size / 4;        // 49152 floats -> 12288 float4
    const int blocks  = (ngroups + 255) / 256;
    attn_bcast_kernel<<<blocks, 256, 0, stream>>>(o, (float4*)d_out, ngroups);
}